// Net_66846870995328
// MI455X (gfx1250) — compile-verified
//
#include <hip/hip_runtime.h>

#define N_NODES  100000
#define N_EDGES  1600000
#define N_GRAPHS 512
#define HDIM     128

typedef float v2f __attribute__((ext_vector_type(2)));
typedef float v8f __attribute__((ext_vector_type(8)));

// ---------------------------------------------------------------------------
// Degree accumulation: one thread per edge, f32 atomics into L2.
// ---------------------------------------------------------------------------
__global__ void deg_kernel(const int* __restrict__ src, const int* __restrict__ dst,
                           float* __restrict__ deg_out, float* __restrict__ deg_in) {
    int e = blockIdx.x * blockDim.x + threadIdx.x;
    if (e < N_EDGES) {
        atomicAdd(&deg_out[src[e]], 1.0f);
        atomicAdd(&deg_in[dst[e]], 1.0f);
    }
}

// deg -> rsqrt(max(deg,1)) in place, over both deg arrays (2N contiguous).
__global__ void norm_kernel(float* __restrict__ deg, int n) {
    int i = blockIdx.x * blockDim.x + threadIdx.x;
    if (i < n) deg[i] = rsqrtf(fmaxf(deg[i], 1.0f));
}

// ---------------------------------------------------------------------------
// Edge scatter: one wave32 per edge; lane moves float4 (32*4 = 128 feats).
// msg = h[src] * norm_out[src], atomically accumulated into agg[dst].
// Whole working set is L2-resident (51 MB << 192 MB L2).
// ---------------------------------------------------------------------------
__global__ void scatter_kernel(const float* __restrict__ h,
                               const int* __restrict__ src, const int* __restrict__ dst,
                               const float* __restrict__ norm_out,
                               float* __restrict__ agg) {
    int e    = (blockIdx.x * blockDim.x + threadIdx.x) >> 5;
    int lane = threadIdx.x & 31;
    if (e >= N_EDGES) return;
    int s = src[e], d = dst[e];
    float ns = norm_out[s];
    float4 m = ((const float4*)(h + (size_t)s * HDIM))[lane];
    float* ap = agg + (size_t)d * HDIM + lane * 4;
    atomicAdd(ap + 0, m.x * ns);
    atomicAdd(ap + 1, m.y * ns);
    atomicAdd(ap + 2, m.z * ns);
    atomicAdd(ap + 3, m.w * ns);
}

// ---------------------------------------------------------------------------
// WMMA GEMM: out[M,128] = act((rowscale * A[M,128]) @ W[128,128] + bias)
// fp32 tensor path: V_WMMA_F32_16X16X4_F32, K=4 per op, 32 ops per tile.
// Block = 256 threads = 8 waves; block covers 16 rows x 128 cols,
// each wave owns one 16x16 tile. M must be a multiple of 16 (100000, 512 are).
//
// f32 16x4 A fragment: lane tl -> row = tl&15, k-pair base = 2*(tl>>4).
// f32 4x16 B fragment: lane tl -> col = tl&15, k-pair base = 2*(tl>>4).
// C/D (8 VGPRs):       vgpr v  -> row = v + 8*(tl>>4), col = tl&15.
// ---------------------------------------------------------------------------
__global__ void wmma_gemm_kernel(const float* __restrict__ A,
                                 const float* __restrict__ rowscale,   // may be null
                                 const float* __restrict__ W,
                                 const float* __restrict__ bias,
                                 float* __restrict__ out,
                                 int relu) {
    int wave    = threadIdx.x >> 5;          // 0..7 -> 16-col tile
    int tl      = threadIdx.x & 31;
    int rowBase = blockIdx.x * 16;
    int colBase = wave * 16;
    int r  = tl & 15;
    int kb = (tl >> 4) << 1;                 // 0 or 2

    int row = rowBase + r;
    const float* Arow = A + (size_t)row * HDIM;
    float scale = rowscale ? rowscale[row] : 1.0f;
    int col = colBase + r;

    v8f c = {};
#pragma unroll
    for (int k0 = 0; k0 < HDIM; k0 += 4) {
        float2 av = *(const float2*)(Arow + k0 + kb);   // contiguous k pair
        v2f a; a.x = av.x * scale; a.y = av.y * scale;
        v2f b;
        b.x = W[(size_t)(k0 + kb)     * HDIM + col];
        b.y = W[(size_t)(k0 + kb + 1) * HDIM + col];
        c = __builtin_amdgcn_wmma_f32_16x16x4_f32(
                /*neg_a=*/false, a, /*neg_b=*/false, b,
                /*c_mod=*/(short)0, c, /*reuse_a=*/false, /*reuse_b=*/false);
    }

    int mhi = (tl >> 4) * 8;
    float bv = bias[col];
#pragma unroll
    for (int v = 0; v < 8; ++v) {
        float val = c[v] + bv;
        if (relu) val = fmaxf(val, 0.0f);
        out[(size_t)(rowBase + v + mhi) * HDIM + col] = val;
    }
}

// ---------------------------------------------------------------------------
// Readout: one wave per node, float4 per lane, atomic add into g[graph].
// ---------------------------------------------------------------------------
__global__ void readout_kernel(const float* __restrict__ h,
                               const int* __restrict__ n2g,
                               float* __restrict__ g) {
    int node = (blockIdx.x * blockDim.x + threadIdx.x) >> 5;
    int lane = threadIdx.x & 31;
    if (node >= N_NODES) return;
    int gi = n2g[node];
    float4 v = ((const float4*)(h + (size_t)node * HDIM))[lane];
    float* gp = g + (size_t)gi * HDIM + lane * 4;
    atomicAdd(gp + 0, v.x);
    atomicAdd(gp + 1, v.y);
    atomicAdd(gp + 2, v.z);
    atomicAdd(gp + 3, v.w);
}

// ---------------------------------------------------------------------------
// Final head: out[g] = t[g,:] . Wm2[:,0] + bm2. 512 dot products of length 128.
// ---------------------------------------------------------------------------
__global__ void head2_kernel(const float* __restrict__ t,
                             const float* __restrict__ Wm2,
                             const float* __restrict__ bm2,
                             float* __restrict__ out) {
    int gi = blockIdx.x * blockDim.x + threadIdx.x;
    if (gi >= N_GRAPHS) return;
    const float4* tp = (const float4*)(t + (size_t)gi * HDIM);
    const float4* wp = (const float4*)Wm2;
    float acc = 0.0f;
#pragma unroll 8
    for (int i = 0; i < HDIM / 4; ++i) {
        float4 a = tp[i];
        float4 w = wp[i];
        acc += a.x * w.x + a.y * w.y + a.z * w.z + a.w * w.w;
    }
    out[gi] = acc + bm2[0];
}

// ---------------------------------------------------------------------------
extern "C" void kernel_launch(void* const* d_in, const int* in_sizes, int n_in,
                              void* d_out, int out_size, void* d_ws, size_t ws_size,
                              hipStream_t stream) {
    const float* x   = (const float*)d_in[0];
    const int*   src = (const int*)  d_in[1];
    const int*   dst = (const int*)  d_in[2];
    const int*   n2g = (const int*)  d_in[3];
    const float* W0  = (const float*)d_in[4];
    const float* b0  = (const float*)d_in[5];
    const float* W1  = (const float*)d_in[6];
    const float* b1  = (const float*)d_in[7];
    const float* W2  = (const float*)d_in[8];
    const float* b2  = (const float*)d_in[9];
    const float* Wm1 = (const float*)d_in[10];
    const float* bm1 = (const float*)d_in[11];
    const float* Wm2 = (const float*)d_in[12];
    const float* bm2 = (const float*)d_in[13];
    float* out = (float*)d_out;

    float* ws = (float*)d_ws;
    size_t off = 0;
    float* norm_out = ws + off; off += N_NODES;
    float* norm_in  = ws + off; off += N_NODES;
    float* bufA     = ws + off; off += (size_t)N_NODES * HDIM;   // agg buffer
    float* bufB     = ws + off; off += (size_t)N_NODES * HDIM;   // h buffer
    float* gbuf     = ws + off; off += (size_t)N_GRAPHS * HDIM;
    float* tbuf     = ws + off; off += (size_t)N_GRAPHS * HDIM;

    // Degree norms (norm_out / norm_in arrays are contiguous -> one memset/map).
    hipMemsetAsync(norm_out, 0, 2 * (size_t)N_NODES * sizeof(float), stream);
    deg_kernel<<<(N_EDGES + 255) / 256, 256, 0, stream>>>(src, dst, norm_out, norm_in);
    norm_kernel<<<(2 * N_NODES + 255) / 256, 256, 0, stream>>>(norm_out, 2 * N_NODES);

    const float* Ws[3] = {W0, W1, W2};
    const float* bs[3] = {b0, b1, b2};
    const float* hcur = x;
    for (int l = 0; l < 3; ++l) {
        hipMemsetAsync(bufA, 0, (size_t)N_NODES * HDIM * sizeof(float), stream);
        scatter_kernel<<<(N_EDGES * 32) / 256, 256, 0, stream>>>(hcur, src, dst, norm_out, bufA);
        wmma_gemm_kernel<<<N_NODES / 16, 256, 0, stream>>>(bufA, norm_in, Ws[l], bs[l], bufB, 1);
        hcur = bufB;
    }

    // Readout + MLP head.
    hipMemsetAsync(gbuf, 0, (size_t)N_GRAPHS * HDIM * sizeof(float), stream);
    readout_kernel<<<(N_NODES * 32) / 256, 256, 0, stream>>>(bufB, n2g, gbuf);
    wmma_gemm_kernel<<<N_GRAPHS / 16, 256, 0, stream>>>(gbuf, nullptr, Wm1, bm1, tbuf, 1);
    head2_kernel<<<(N_GRAPHS + 255) / 256, 256, 0, stream>>>(tbuf, Wm2, bm2, out);
}